// TypeSpecificProjection_90108413870254
// MI455X (gfx1250) — compile-verified
//
#include <hip/hip_runtime.h>
#include <stdint.h>

typedef __attribute__((ext_vector_type(2))) float v2f;
typedef __attribute__((ext_vector_type(8))) float v8f;

#define OUTD 256
#define HALF 128
#define WSTR 136   // weight LDS row stride (floats): 2*WSTR % 64 == 16 -> conflict-free halves
#define FSTR 132   // feature LDS row stride (floats): full wave hits all 64 banks exactly once

// LDS layout (float offsets)
#define W_OFF 0                        // 320 K-rows * WSTR
#define B_OFF (320 * WSTR)             // 43520 : 4*128 bias
#define F0_OFF (B_OFF + 4 * HALF)      // 44032 : feature buffer 0 (32 rows)
#define FSZ (32 * FSTR)                // 4224
#define F1_OFF (F0_OFF + FSZ)          // 48256 : feature buffer 1
#define T0_OFF (F1_OFF + FSZ)          // 52480 : 32 type ids (buf 0)
#define T1_OFF (T0_OFF + 32)           // 52512 : 32 type ids (buf 1)
#define SMEM_FLOATS (T1_OFF + 32)      // 52544 floats = 210176 bytes

#ifndef __has_builtin
#define __has_builtin(x) 0
#endif

#if __has_builtin(__builtin_amdgcn_global_load_async_to_lds_b128) && \
    __has_builtin(__builtin_amdgcn_global_load_async_to_lds_b32)
#define USE_ASYNC 1
#else
#define USE_ASYNC 0
#endif

#if __has_builtin(__builtin_amdgcn_s_wait_asynccnt)
#define WAIT_ASYNCCNT(n) __builtin_amdgcn_s_wait_asynccnt(n)
#else
#define WAIT_ASYNCCNT(n) asm volatile("s_wait_asynccnt %0" ::"n"(n) : "memory")
#endif

#if USE_ASYNC
// Builtin prototypes (from compiler diagnostics):
//   b128: (AS1 int4*, AS3 int4*, Ii offset, Ii cpol)
//   b32 : (AS1 int*,  AS3 int*,  Ii offset, Ii cpol)
typedef int v4i_t __attribute__((vector_size(16)));
typedef __attribute__((address_space(1))) v4i_t gv4i;
typedef __attribute__((address_space(3))) v4i_t lv4i;
typedef __attribute__((address_space(1))) int gint;
typedef __attribute__((address_space(3))) int lint;

// Flat LDS addresses carry the LDS byte offset in their low 32 bits (ISA 10.2),
// so a 32-bit truncation yields the AS3 pointer.
__device__ __forceinline__ void async_cp_b128(const float* gsrc, const float* ldst) {
  __builtin_amdgcn_global_load_async_to_lds_b128(
      (gv4i*)(uintptr_t)gsrc, (lv4i*)(uint32_t)(uintptr_t)ldst, 0, 0);
}
__device__ __forceinline__ void async_cp_b32(const int* gsrc, const int* ldst) {
  __builtin_amdgcn_global_load_async_to_lds_b32(
      (gint*)(uintptr_t)gsrc, (lint*)(uint32_t)(uintptr_t)ldst, 0, 0);
}
#endif

__device__ __forceinline__ float4 ld4(const float* p) { return *(const float4*)p; }

// One (type, 16-col tile) projection over a staged 16-row sub-tile.
template<int T, int D, int WROW>
__device__ __forceinline__ void project_type(
    const float* sm, const float* fl, const int* tl, float* __restrict__ out,
    int rowbase, int n, int colbase, int w, int lo, int hi)
{
  const float* wl = sm + W_OFF;
  const float* bl = sm + B_OFF;

  const int colw = w * 16 + lo;
  const float bv = bl[T * HALF + colw];
  v8f acc = {bv, bv, bv, bv, bv, bv, bv, bv};

#pragma unroll
  for (int k = 0; k < D; k += 4) {
    // A tile 16x4 f32: lane holds M=lo, K = {k+2*hi, k+1+2*hi}
    v2f a = *(const v2f*)&fl[lo * FSTR + k + 2 * hi];
    // B tile 4x16 f32: lane holds N=colw, K = {k+2*hi, k+1+2*hi}
    const float* wp = &wl[(WROW + k + 2 * hi) * WSTR + colw];
    v2f b;
    b.x = wp[0];
    b.y = wp[WSTR];
    acc = __builtin_amdgcn_wmma_f32_16x16x4_f32(
        false, a, false, b, (short)0, acc, false, false);
  }

  const int col = colbase + colw;
#pragma unroll
  for (int v = 0; v < 8; ++v) {
    const int r = v + 8 * hi;          // row within 16-row sub-tile held by this vgpr
    const int row = rowbase + r;
    if (tl[r] == T && row < n)
      out[(size_t)row * OUTD + col] = acc[v];
  }
}

__device__ __forceinline__ void compute16(
    const float* sm, const float* fl, const int* tl, float* __restrict__ out,
    int rowbase, int n, int colbase, int w, int lo, int hi)
{
  project_type<0,  64,   0>(sm, fl, tl, out, rowbase, n, colbase, w, lo, hi);
  project_type<1, 128,  64>(sm, fl, tl, out, rowbase, n, colbase, w, lo, hi);
  project_type<2,  96, 192>(sm, fl, tl, out, rowbase, n, colbase, w, lo, hi);
  project_type<3,  32, 288>(sm, fl, tl, out, rowbase, n, colbase, w, lo, hi);
}

__global__ __launch_bounds__(256)
void TypeSpecificProjection_kernel(
    const float* __restrict__ feat, const int* __restrict__ typ,
    const float* __restrict__ W0, const float* __restrict__ b0,
    const float* __restrict__ W1, const float* __restrict__ b1,
    const float* __restrict__ W2, const float* __restrict__ b2,
    const float* __restrict__ W3, const float* __restrict__ b3,
    float* __restrict__ out, int n)
{
  extern __shared__ float sm[];
  const int tid = threadIdx.x;
  const int h = blockIdx.x & 1;          // column half: cols [h*128, h*128+128)
  const int rwg = blockIdx.x >> 1;       // row partition id
  const int nrwg = gridDim.x >> 1;
  const int colbase = h * HALF;

  // ---- one-time weight + bias staging into LDS (rows padded to WSTR, 170 KB) ----
  const float* Wg[4] = {W0, W1, W2, W3};
  const float* bg[4] = {b0, b1, b2, b3};
  const int DT[4] = {64, 128, 96, 32};
  const int WB[4] = {0, 64, 192, 288};   // K-row base of each type slab
#pragma unroll
  for (int t = 0; t < 4; ++t) {
    const float* W = Wg[t];
    const int nq = DT[t] * (HALF / 4);   // float4 slots
    for (int i = tid; i < nq; i += 256) {
      const int r = i >> 5;              // 32 float4 per 128-col row
      const int c = (i & 31) << 2;
      float4 v = ld4(W + (size_t)r * OUTD + colbase + c);
      *(float4*)&sm[W_OFF + (WB[t] + r) * WSTR + c] = v;
    }
  }
  for (int i = tid; i < 4 * HALF; i += 256) {
    const int t = i >> 7, c = i & (HALF - 1);
    sm[B_OFF + i] = bg[t][colbase + c];
  }
  __syncthreads();

  const int lane = tid & 31;
  const int w = tid >> 5;                // wave id -> 16-col tile within half
  const int lo = lane & 15;
  const int hi = lane >> 4;
  const int ntiles = (n + 31) >> 5;      // 32-row tiles

#if USE_ASYNC
  // ---- double-buffered async pipeline: 5 async ops/wave/tile (4x b128 + 1x b32) ----
  auto stage = [&](int buf, int tile) {
    const int rb = tile << 5;
    float* fb = sm + (buf ? F1_OFF : F0_OFF);
    int* tb = (int*)(sm + (buf ? T1_OFF : T0_OFF));
    for (int i = tid; i < 1024; i += 256) {    // 1024 float4s, 4 per thread
      const int r = i >> 5;
      const int c = (i & 31) << 2;
      const int gr = rb + r;
      if (gr < n) async_cp_b128(feat + (size_t)gr * HALF + c, &fb[r * FSTR + c]);
    }
    const int gr = rb + lane;                  // every wave stages types (uniform count)
    if (gr < n) async_cp_b32(typ + gr, &tb[lane]);
  };

  int tile = rwg;
  int buf = 0;
  if (tile < ntiles) stage(0, tile);
  for (; tile < ntiles; tile += nrwg) {
    const int nt = tile + nrwg;
    const bool has_next = (nt < ntiles);
    if (has_next) stage(buf ^ 1, nt);
    // counted wait: allow the 5 just-issued prefetch ops to stay in flight
    if (has_next && (((nt << 5) + 32) <= n)) {
      WAIT_ASYNCCNT(5);
    } else {
      WAIT_ASYNCCNT(0);
    }
    __syncthreads();

    const int rb = tile << 5;
    const float* fl = sm + (buf ? F1_OFF : F0_OFF);
    const int* tl = (const int*)(sm + (buf ? T1_OFF : T0_OFF));
    compute16(sm, fl,             tl,      out, rb,      n, colbase, w, lo, hi);
    compute16(sm, fl + 16 * FSTR, tl + 16, out, rb + 16, n, colbase, w, lo, hi);
    __syncthreads();               // WAR: all reads done before next restage
    buf ^= 1;
  }
#else
  // ---- fallback: synchronous single-buffer staging ----
  for (int tile = rwg; tile < ntiles; tile += nrwg) {
    const int rb = tile << 5;
    for (int i = tid; i < 1024; i += 256) {
      const int r = i >> 5;
      const int c = (i & 31) << 2;
      const int gr = rb + r;
      float4 v = {0.f, 0.f, 0.f, 0.f};
      if (gr < n) v = ld4(feat + (size_t)gr * HALF + c);
      *(float4*)&sm[F0_OFF + r * FSTR + c] = v;
    }
    if (tid < 32)
      ((int*)(sm + T0_OFF))[tid] = (rb + tid < n) ? typ[rb + tid] : -1;
    __syncthreads();

    const float* fl = sm + F0_OFF;
    const int* tl = (const int*)(sm + T0_OFF);
    compute16(sm, fl,             tl,      out, rb,      n, colbase, w, lo, hi);
    compute16(sm, fl + 16 * FSTR, tl + 16, out, rb + 16, n, colbase, w, lo, hi);
    __syncthreads();
  }
#endif
}

extern "C" void kernel_launch(void* const* d_in, const int* in_sizes, int n_in,
                              void* d_out, int out_size, void* d_ws, size_t ws_size,
                              hipStream_t stream) {
  const float* feat = (const float*)d_in[0];
  const int*   typ  = (const int*)d_in[1];
  const float* W0 = (const float*)d_in[2];
  const float* b0 = (const float*)d_in[3];
  const float* W1 = (const float*)d_in[4];
  const float* b1 = (const float*)d_in[5];
  const float* W2 = (const float*)d_in[6];
  const float* b2 = (const float*)d_in[7];
  const float* W3 = (const float*)d_in[8];
  const float* b3 = (const float*)d_in[9];
  float* out = (float*)d_out;
  const int n = in_sizes[1];             // N_VERTS

  const size_t smem = (size_t)SMEM_FLOATS * sizeof(float);
  // Allow >default dynamic LDS (210,176 B; hardware permits 320 KB/WG).
  (void)hipFuncSetAttribute((const void*)TypeSpecificProjection_kernel,
                            hipFuncAttributeMaxDynamicSharedMemorySize, (int)smem);

  const int blocks = 2 * 512;            // 2 column halves x 512 row partitions
  TypeSpecificProjection_kernel<<<dim3(blocks), dim3(256), smem, stream>>>(
      feat, typ, W0, b0, W1, b1, W2, b2, W3, b3, out, n);
}